// MambaSpikingOutput_41240275976485
// MI455X (gfx1250) — compile-verified
//
#include <hip/hip_runtime.h>
#include <cstdint>

// Problem constants (match reference)
#define LIF_B 8
#define LIF_S 2048
#define LIF_H 4096
#define REFRACTORY 5.0f

// Pipeline constants
#define TT    32                        // timesteps per LDS tile
#define NBUF  4                         // ring-buffer depth (16 KB/wave in flight)
#define NT    (LIF_S / TT)              // 64 tiles per channel-group
#define WPB   2                         // waves per block
#define NWAVE ((LIF_B * LIF_H) / 32)    // 1024 waves total
#define NBLK  (NWAVE / WPB)             // 512 blocks

// Split-counter waits (CDNA5). Token-pasted immediates -> SOPP simm16.
#define WAIT_ASYNCCNT(imm) asm volatile("s_wait_asynccnt " #imm ::: "memory")
#define WAIT_DSCNT0()      asm volatile("s_wait_dscnt 0x0" ::: "memory")

// Issue one 32x32-float tile (4 KB) as 8 async b128 DMA ops: each op moves
// 32 lanes x 16 B = 4 rows of the tile. GVS addressing: uniform SGPR base +
// per-lane signed 32-bit VGPR offset. Tracked by ASYNCcnt.
__device__ __forceinline__ void issue_tile_async(uint64_t gbase,
                                                 unsigned gv_tile_lane,
                                                 unsigned lds_buf_lane) {
#pragma unroll
  for (int j = 0; j < TT / 4; ++j) {
    unsigned ga = gv_tile_lane + (unsigned)j * (4u * LIF_H * 4u);  // +4 rows in global
    unsigned la = lds_buf_lane + (unsigned)j * (4u * 32u * 4u);    // +4 rows in LDS
    asm volatile("global_load_async_to_lds_b128 %0, %1, %2"
                 :
                 : "v"(la), "v"(ga), "s"(gbase)
                 : "memory");
  }
}

// Consume one tile from LDS: 32 sequential LIF steps for this lane's channel.
__device__ __forceinline__ void compute_tile(const float* __restrict__ tile,
                                             int lane,
                                             float* __restrict__ outp,
                                             float& mem, float& ref,
                                             float thr, float dec) {
#pragma unroll
  for (int t = 0; t < TT; ++t) {
    float v = tile[t * 32 + lane];          // ds_load_b32, bank-conflict free
    mem = dec * mem + v;                    // leaky integrate
    ref = fmaxf(ref - 1.0f, 0.0f);          // refractory countdown
    bool cs   = (ref == 0.0f) && (mem > thr);
    float spk = cs ? 1.0f : 0.0f;
    mem = cs ? 0.0f : mem;                  // reset on spike
    ref = cs ? (ref + REFRACTORY) : ref;    // enter refractory
    outp[(size_t)t * LIF_H] = spk;          // global_store_b32 (128 B/wave)
  }
}

__launch_bounds__(WPB * 32)
__global__ void lif_scan_kernel(const float* __restrict__ x,
                                const float* __restrict__ thr_p,
                                const float* __restrict__ dec_p,
                                float* __restrict__ out) {
  __shared__ __attribute__((aligned(128))) float lds[WPB][NBUF][TT][32];

  const int lane = (int)(threadIdx.x & 31u);
  const int wv   = (int)(threadIdx.x >> 5);
  const int w    = (int)blockIdx.x * WPB + wv;   // global wave id: 0..1023
  const int c0   = w * 32;                       // first channel of this wave
  const int b    = c0 / LIF_H;
  const int h0   = c0 % LIF_H;

  const float thr = thr_p[0];
  const float dec = dec_p[0];

  // Uniform 64-bit base for this wave's channel group (start of t=0 row).
  const uint64_t gbase =
      (uint64_t)(uintptr_t)(x + ((size_t)b * LIF_S) * LIF_H + (size_t)h0);

  // Per-lane layout inside one async op: 4 rows x 8 cols of 16 B.
  const int r  = lane >> 3;
  const int cc = lane & 7;
  const unsigned gv_lane  = (unsigned)(r * (LIF_H * 4) + cc * 16);
  const unsigned lds_lane = (unsigned)(r * (32 * 4) + cc * 16);
  // Flat shared pointer low 32 bits == wave-relative LDS byte address (ISA 10.2).
  const unsigned lds_wave = (unsigned)(uintptr_t)(&lds[wv][0][0][0]);

  // Prologue: fill the ring (4 tiles x 8 asyncs = 32 outstanding, <= 63 max).
#pragma unroll
  for (int p = 0; p < NBUF; ++p) {
    issue_tile_async(gbase,
                     gv_lane + (unsigned)p * (TT * LIF_H * 4u),
                     lds_wave + lds_lane + (unsigned)p * (TT * 32u * 4u));
  }

  float mem = 0.0f, ref = 0.0f;
  float* outb = out + ((size_t)b * LIF_S) * LIF_H + (size_t)h0 + (size_t)lane;

  // Steady state: tile i resident when ASYNCcnt <= 3 tiles * 8 = 24 (in-order).
  for (int i = 0; i < NT - NBUF; ++i) {
    WAIT_ASYNCCNT(24);
    const float* tile = &lds[wv][i & (NBUF - 1)][0][0];
    compute_tile(tile, lane, outb + (size_t)i * TT * LIF_H, mem, ref, thr, dec);
    WAIT_DSCNT0();  // all ds reads of this buffer done before DMA overwrites it
    issue_tile_async(gbase,
                     gv_lane + (unsigned)(i + NBUF) * (TT * LIF_H * 4u),
                     lds_wave + lds_lane +
                         (unsigned)(i & (NBUF - 1)) * (TT * 32u * 4u));
  }

  // Drain: ASYNCcnt completes in order, so decreasing immediates pin each tile.
  WAIT_ASYNCCNT(24);
  compute_tile(&lds[wv][(NT - 4) & (NBUF - 1)][0][0], lane,
               outb + (size_t)(NT - 4) * TT * LIF_H, mem, ref, thr, dec);
  WAIT_ASYNCCNT(16);
  compute_tile(&lds[wv][(NT - 3) & (NBUF - 1)][0][0], lane,
               outb + (size_t)(NT - 3) * TT * LIF_H, mem, ref, thr, dec);
  WAIT_ASYNCCNT(8);
  compute_tile(&lds[wv][(NT - 2) & (NBUF - 1)][0][0], lane,
               outb + (size_t)(NT - 2) * TT * LIF_H, mem, ref, thr, dec);
  WAIT_ASYNCCNT(0);
  compute_tile(&lds[wv][(NT - 1) & (NBUF - 1)][0][0], lane,
               outb + (size_t)(NT - 1) * TT * LIF_H, mem, ref, thr, dec);
}

extern "C" void kernel_launch(void* const* d_in, const int* in_sizes, int n_in,
                              void* d_out, int out_size, void* d_ws, size_t ws_size,
                              hipStream_t stream) {
  (void)in_sizes; (void)n_in; (void)out_size; (void)d_ws; (void)ws_size;
  const float* x     = (const float*)d_in[0];
  const float* thr_p = (const float*)d_in[1];  // single-element array
  const float* dec_p = (const float*)d_in[2];  // single-element array
  float* out = (float*)d_out;

  lif_scan_kernel<<<NBLK, WPB * 32, 0, stream>>>(x, thr_p, dec_p, out);
}